// Model_41266045780525
// MI455X (gfx1250) — compile-verified
//
#include <hip/hip_runtime.h>
#include <hip/hip_bf16.h>
#include <math.h>

// Problem constants (from reference)
#define B_   8
#define C_   32
#define IND  32
#define INH  128
#define INW  128
#define OD   4
#define OH   8
#define OW   8
#define S_   (OD*OH*OW)   // 256

typedef __attribute__((ext_vector_type(2))) float v2f;
typedef __attribute__((ext_vector_type(8))) float v8f;

// Reflection-pad + adaptive-pool windows collapse to clipped source windows.
// H/W (128 -> 8, padded 130):  s = max(floor(i*130/8)-1, 0), e = min(ceil((i+1)*130/8)-1, 128)
__device__ __forceinline__ void hwin(int i, int& s, int& e) {
    int sp = (i * 130) >> 3;
    int ep = (130 * (i + 1) + 7) >> 3;
    s = (sp > 0) ? sp - 1 : 0;
    e = (ep - 1 > INW) ? INW : ep - 1;
}
// D (32 -> 4, padded 34)
__device__ __forceinline__ void dwin(int i, int& s, int& e) {
    int sp = (i * 34) >> 2;
    int ep = (34 * (i + 1) + 3) >> 2;
    s = (sp > 0) ? sp - 1 : 0;
    e = (ep - 1 > IND) ? IND : ep - 1;
}

// ---------------------------------------------------------------------------
// Kernel 1: per-(b,c,d) plane, pool H and W windows.
// Reads each input element exactly once (coalesced float4), writes 64 floats.
// out1[((b*C + c)*IND + d)*64 + oh*8 + ow]
// ---------------------------------------------------------------------------
__global__ void __launch_bounds__(256) pool_hw_kernel(const float* __restrict__ x,
                                                      float* __restrict__ out1) {
    __shared__ float plane[INH * INW];   // 64 KB
    __shared__ float part[256];

    const int blk = blockIdx.x;          // (b*C + c)*IND + d
    const int t   = threadIdx.x;

    // Coalesced load of the whole 128x128 plane into LDS.
    const float4* src = (const float4*)(x + (size_t)blk * (INH * INW));
    float4* dst = (float4*)plane;
#pragma unroll
    for (int i = 0; i < (INH * INW / 4) / 256; ++i)   // 16 iters
        dst[i * 256 + t] = src[i * 256 + t];
    __syncthreads();

    // thread -> (replica r, oh, ow); replica r handles rows h = hs+r, hs+r+4, ...
    const int ow = t & 7;
    const int oh = (t >> 3) & 7;
    const int r  = t >> 6;

    int hs, he, ws, we;
    hwin(oh, hs, he);
    hwin(ow, ws, we);

    float m = -__builtin_inff();
    for (int h = hs + r; h < he; h += 4) {
        const float* row = plane + h * INW;
        for (int w = ws; w < we; ++w)
            m = fmaxf(m, row[w]);
    }
    part[t] = m;
    __syncthreads();

    if (t < 64) {
        float v = fmaxf(fmaxf(part[t], part[t + 64]),
                        fmaxf(part[t + 128], part[t + 192]));
        out1[(size_t)blk * 64 + t] = v;
    }
}

// ---------------------------------------------------------------------------
// Kernel 2: one block per batch. D-pool, then the attention head.
// x_proj (S=256 x 32) = x_feat (256x32) @ W_proj^T via v_wmma_f32_16x16x4_f32.
// ---------------------------------------------------------------------------
__global__ void __launch_bounds__(256) attn_kernel(const float* __restrict__ out1,
                                                   const float* __restrict__ Wp,
                                                   const float* __restrict__ Wf,
                                                   const float* __restrict__ bf,
                                                   float* __restrict__ outp) {
    __shared__ float p_lds[C_ * S_];   // [c][s]  32 KB
    __shared__ float xp[S_ * C_];      // [s][d]  32 KB
    __shared__ float ctx[C_];
    __shared__ float att[S_];
    __shared__ float red[S_];
    __shared__ float wgt[C_];
    __shared__ float cl[C_];

    const int b = blockIdx.x;
    const int t = threadIdx.x;

    // ---- Step 1: max over D windows -> p_lds[c][s], s = od*64 + oh*8 + ow
    const float* base = out1 + (size_t)b * C_ * IND * 64;
    for (int it = 0; it < (C_ * S_) / 256; ++it) {   // 32 iters
        int o   = it * 256 + t;
        int c   = o >> 8;
        int rem = o & 255;                // = s
        int od  = rem >> 6;
        int j   = rem & 63;
        int ds, de;
        dwin(od, ds, de);
        float m = -__builtin_inff();
        for (int d = ds; d < de; ++d)
            m = fmaxf(m, base[(c * IND + d) * 64 + j]);
        p_lds[c * S_ + rem] = m;
    }
    __syncthreads();

    // ---- Step 2: x_proj[s][d] = sum_c x_feat[s][c] * Wp[d][c]  (WMMA f32 16x16x4)
    // 16 s-tiles x 2 d-tiles = 32 tiles; 8 waves x 4 tiles each. EXEC all ones.
    {
        const int wv   = t >> 5;
        const int lane = t & 31;
        const int lr   = lane & 15;
        const bool hi  = lane >= 16;
        for (int q = 0; q < 4; ++q) {
            const int tile = wv * 4 + q;
            const int s0 = (tile >> 1) * 16;
            const int d0 = (tile & 1) * 16;
            v8f acc = {0.f, 0.f, 0.f, 0.f, 0.f, 0.f, 0.f, 0.f};
#pragma unroll
            for (int kk = 0; kk < 8; ++kk) {
                // ISA f32 A(16x4) layout: VGPR0 = K=0 (lanes 0-15) / K=2 (lanes 16-31),
                // VGPR1 = K=1 / K=3. B(4x16) mirrors this with N across lanes.
                const int k0 = kk * 4 + (hi ? 2 : 0);
                v2f a, bb;
                a.x  = p_lds[(k0 + 0) * S_ + s0 + lr];   // x_feat[s0+lr][k0]
                a.y  = p_lds[(k0 + 1) * S_ + s0 + lr];
                bb.x = Wp[(d0 + lr) * C_ + k0 + 0];      // B[k][n] = Wp[n][k]
                bb.y = Wp[(d0 + lr) * C_ + k0 + 1];
                acc = __builtin_amdgcn_wmma_f32_16x16x4_f32(
                        /*neg_a=*/false, a, /*neg_b=*/false, bb,
                        /*c_mod=*/(short)0, acc, /*reuse_a=*/false, /*reuse_b=*/false);
            }
            // C/D layout: VGPR j -> M = j (lanes 0-15) / 8+j (lanes 16-31), N = lane%16
#pragma unroll
            for (int j = 0; j < 8; ++j) {
                const int row = s0 + (hi ? 8 + j : j);
                xp[row * C_ + d0 + lr] = acc[j];
            }
        }
    }
    __syncthreads();

    // ---- Step 3: context[d] = mean_s x_proj[s][d]
    if (t < C_) {
        float s = 0.f;
        for (int si = 0; si < S_; ++si) s += xp[si * C_ + t];
        ctx[t] = s * (1.0f / S_);
    }
    __syncthreads();

    // ---- Step 4: att_logits[s] = ctx . x_proj[s], softmax over s
    float l = 0.f;
    for (int d = 0; d < C_; ++d) l += ctx[d] * xp[t * C_ + d];
    red[t] = l;
    __syncthreads();
    for (int str = 128; str > 0; str >>= 1) {
        if (t < str) red[t] = fmaxf(red[t], red[t + str]);
        __syncthreads();
    }
    const float mx = red[0];
    __syncthreads();
    const float e = expf(l - mx);
    red[t] = e;
    __syncthreads();
    for (int str = 128; str > 0; str >>= 1) {
        if (t < str) red[t] += red[t + str];
        __syncthreads();
    }
    att[t] = e / red[0];
    __syncthreads();

    // ---- Step 5: weighted[d] = sum_s att[s] * x_proj[s][d]
    if (t < C_) {
        float s = 0.f;
        for (int si = 0; si < S_; ++si) s += att[si] * xp[si * C_ + t];
        wgt[t] = s;
    }
    __syncthreads();

    // ---- Step 6: channel_logits = weighted @ Wf^T + bf ; log_softmax
    if (t < C_) {
        float s = bf[t];
        for (int d = 0; d < C_; ++d) s += wgt[d] * Wf[t * C_ + d];
        cl[t] = s;
    }
    __syncthreads();
    if (t == 0) {
        float m2 = -__builtin_inff();
        for (int i = 0; i < C_; ++i) m2 = fmaxf(m2, cl[i]);
        float sm = 0.f;
        for (int i = 0; i < C_; ++i) sm += expf(cl[i] - m2);
        red[0] = m2 + logf(sm);
    }
    __syncthreads();
    if (t < C_) outp[b * C_ + t] = cl[t] - red[0];
}

// ---------------------------------------------------------------------------
extern "C" void kernel_launch(void* const* d_in, const int* in_sizes, int n_in,
                              void* d_out, int out_size, void* d_ws, size_t ws_size,
                              hipStream_t stream) {
    const float* x   = (const float*)d_in[0];   // (8,32,32,128,128)
    const float* Wp  = (const float*)d_in[1];   // (32,32)
    const float* Wf  = (const float*)d_in[2];   // (32,32)
    const float* bfi = (const float*)d_in[3];   // (32,)
    float* out  = (float*)d_out;                // (8,32)
    float* out1 = (float*)d_ws;                 // B*C*IND*64 floats = 2 MB scratch

    pool_hw_kernel<<<B_ * C_ * IND, 256, 0, stream>>>(x, out1);
    attn_kernel<<<B_, 256, 0, stream>>>(out1, Wp, Wf, bfi, out);
}